// SAGPool_10986526343677
// MI455X (gfx1250) — compile-verified
//
#include <hip/hip_runtime.h>
#include <hip/hip_bf16.h>
#include <math.h>

typedef float v2f __attribute__((ext_vector_type(2)));
typedef float v8f __attribute__((ext_vector_type(8)));

#define N_NODES 8192
#define N_FEAT  256
#define K_SEL   4096

// ---------- 1) t = X @ kernel : one block per row ----------
__global__ __launch_bounds__(256) void xk_kernel(const float* __restrict__ X,
                                                 const float* __restrict__ kern,
                                                 float* __restrict__ t) {
  __shared__ float red[256];
  const int row = blockIdx.x, tid = threadIdx.x;
  red[tid] = X[(size_t)row * N_FEAT + tid] * kern[tid];
  __syncthreads();
  for (int s = 128; s > 0; s >>= 1) {
    if (tid < s) red[tid] += red[tid + s];
    __syncthreads();
  }
  if (tid == 0) t[row] = red[0];
}

// ---------- 2) y = A @ t : one block per row (A read once, 256 MB) ----------
__global__ __launch_bounds__(256) void av_kernel(const float* __restrict__ A,
                                                 const float* __restrict__ t,
                                                 float* __restrict__ y) {
  __shared__ float red[256];
  const int row = blockIdx.x, tid = threadIdx.x;
  const float* ar = A + (size_t)row * N_NODES;
  float s = 0.f;
  for (int j = tid; j < N_NODES; j += 256) s += ar[j] * t[j];
  red[tid] = s;
  __syncthreads();
  for (int k = 128; k > 0; k >>= 1) {
    if (tid < k) red[tid] += red[tid + k];
    __syncthreads();
  }
  if (tid == 0) y[row] = red[0];
}

// ---------- 3) rank-based top-K selection (K = N/2), scores in LDS ----------
__global__ __launch_bounds__(256) void rank_kernel(const float* __restrict__ y,
                                                   int* __restrict__ sel) {
  __shared__ float sy[N_NODES];           // 32 KB
  const int tid = threadIdx.x;
  for (int j = tid; j < N_NODES; j += 256) sy[j] = y[j];
  __syncthreads();
  const int i = blockIdx.x * 256 + tid;
  const float yi = sy[i];
  int cnt = 0;
  for (int j = 0; j < N_NODES; ++j) {
    const float yj = sy[j];
    cnt += (yj > yi) || (yj == yi && j < i);   // strict total order (jax tie-break)
  }
  sel[i] = (cnt < K_SEL) ? 1 : 0;
}

// ---------- 4) prefix-sum compaction -> sorted-ascending idx ----------
__global__ __launch_bounds__(1024) void scan_kernel(const int* __restrict__ sel,
                                                    int* __restrict__ idxb) {
  __shared__ int sums[1024];
  const int tid = threadIdx.x;
  const int base = tid * 8;
  int loc[8]; int s = 0;
#pragma unroll
  for (int i = 0; i < 8; ++i) { loc[i] = s; s += sel[base + i]; }
  sums[tid] = s;
  __syncthreads();
  for (int off = 1; off < 1024; off <<= 1) {
    int v = 0;
    if (tid >= off) v = sums[tid - off];
    __syncthreads();
    sums[tid] += v;
    __syncthreads();
  }
  const int excl = (tid == 0) ? 0 : sums[tid - 1];
#pragma unroll
  for (int i = 0; i < 8; ++i)
    if (sel[base + i]) idxb[excl + loc[i]] = base + i;
}

// ---------- 5) X_pooled = (X * tanh(y))[idx] ----------
__global__ __launch_bounds__(256) void xpool_kernel(const float* __restrict__ X,
                                                    const float* __restrict__ y,
                                                    const int* __restrict__ idxb,
                                                    float* __restrict__ outX) {
  const int k = blockIdx.x, tid = threadIdx.x;
  const int r = idxb[k];
  const float g = tanhf(y[r]);
  outX[(size_t)k * N_FEAT + tid] = X[(size_t)r * N_FEAT + tid] * g;
}

// ---------- 6) A_pooled = A[idx,:] @ A[:,idx]  (f32 WMMA GEMM) ----------
// 128x64x16 tiles, 256 threads (8 wave32). Wave w owns M-subtile w (16 rows)
// across the whole 64-col N-tile: per kk-step 1 A ds_load_b64 + 4 B ds_load_b64
// + 4 v_wmma. B tile stored pair-interleaved Bs[kpair][n][2] so a fragment is
// one aligned v2f; pitch 160 puts the kpair+1 half-wave 32 banks away.
#define TM 128
#define TN 64
#define TK 16
#define AP 18            // As pitch: 18*M mod 64 distinct for M=0..15
#define BP2 (2*TN + 32)  // 160: pair-interleaved Bs pitch, conflict-free halves

__global__ __launch_bounds__(256) void apool_gemm(const float* __restrict__ A,
                                                  const int* __restrict__ idxb,
                                                  float* __restrict__ Cout) {
  __shared__ float As[TM * AP];          // 9216 B
  __shared__ float Bs[(TK / 2) * BP2];   // 5120 B
  __shared__ int rows[TM];
  __shared__ int cols[TN];

  const int tid = threadIdx.x;
  const int bm = blockIdx.y * TM;
  const int bn = blockIdx.x * TN;

  if (tid < TM)                rows[tid]      = idxb[bm + tid];
  else if (tid < TM + TN)      cols[tid - TM] = idxb[bn + (tid - TM)];
  __syncthreads();

  const int lane = tid & 31;
  const int wave = tid >> 5;          // 0..7
  const int ll   = lane & 15;
  const int lh   = lane >> 4;         // half-wave: K-pair select / M+8 rows
  const int mt   = wave * 16;         // wave's M subtile (covers all 128 rows)

  v8f c0 = {}, c1 = {}, c2 = {}, c3 = {};

  // A-tile loader: 128x16 floats, 8 per thread (two float4), coalesced row-gather
  const int lm = tid >> 1;            // 0..127
  const int lk = (tid & 1) * 8;       // 0 or 8
  const float* asrc = A + (size_t)rows[lm] * N_NODES + lk;

  // B-tile loader: 16x64 column gather, 4 per thread
  const int be  = tid * 4;
  const int bk  = be >> 6;            // 0..15
  const int bn0 = be & 63;            // 0,4,..,60

  for (int k0 = 0; k0 < N_NODES; k0 += TK) {
    float4 av0 = *(const float4*)(asrc + k0);
    float4 av1 = *(const float4*)(asrc + k0 + 4);
    As[lm * AP + lk + 0] = av0.x;
    As[lm * AP + lk + 1] = av0.y;
    As[lm * AP + lk + 2] = av0.z;
    As[lm * AP + lk + 3] = av0.w;
    As[lm * AP + lk + 4] = av1.x;
    As[lm * AP + lk + 5] = av1.y;
    As[lm * AP + lk + 6] = av1.z;
    As[lm * AP + lk + 7] = av1.w;

    const float* brow = A + (size_t)(k0 + bk) * N_NODES;
#pragma unroll
    for (int i = 0; i < 4; ++i)
      Bs[(bk >> 1) * BP2 + 2 * (bn0 + i) + (bk & 1)] = brow[cols[bn0 + i]];
    __syncthreads();

#pragma unroll
    for (int kk = 0; kk < 4; ++kk) {
      const int kb = kk * 4 + lh * 2;                 // ISA f32 A/B lane layout
      v2f a = *(const v2f*)&As[(mt + ll) * AP + kb];
      const float* bb = &Bs[(kb >> 1) * BP2 + 2 * ll];
      v2f b0 = *(const v2f*)(bb + 0);
      v2f b1 = *(const v2f*)(bb + 32);
      v2f b2 = *(const v2f*)(bb + 64);
      v2f b3 = *(const v2f*)(bb + 96);
      c0 = __builtin_amdgcn_wmma_f32_16x16x4_f32(false, a, false, b0, (short)0, c0, false, false);
      c1 = __builtin_amdgcn_wmma_f32_16x16x4_f32(false, a, false, b1, (short)0, c1, false, false);
      c2 = __builtin_amdgcn_wmma_f32_16x16x4_f32(false, a, false, b2, (short)0, c2, false, false);
      c3 = __builtin_amdgcn_wmma_f32_16x16x4_f32(false, a, false, b3, (short)0, c3, false, false);
    }
    __syncthreads();
  }

  // C layout: VGPR v holds M = v (lanes 0-15) / M = 8+v (lanes 16-31), N = lane%16
#pragma unroll
  for (int v = 0; v < 8; ++v) {
    const int rl = v + lh * 8;
    float* crow = Cout + (size_t)(bm + mt + rl) * K_SEL + bn;
    crow[ 0 + ll] = c0[v];
    crow[16 + ll] = c1[v];
    crow[32 + ll] = c2[v];
    crow[48 + ll] = c3[v];
  }
}

extern "C" void kernel_launch(void* const* d_in, const int* in_sizes, int n_in,
                              void* d_out, int out_size, void* d_ws, size_t ws_size,
                              hipStream_t stream) {
  const float* X    = (const float*)d_in[0];
  const float* A    = (const float*)d_in[1];
  const float* kern = (const float*)d_in[2];

  float* t    = (float*)d_ws;           // 8192
  float* y    = t + N_NODES;            // 8192
  int*   sel  = (int*)(y + N_NODES);    // 8192
  int*   idxb = sel + N_NODES;          // 4096   (total ws use: 112 KB)

  float* outX = (float*)d_out;                       // 4096 x 256
  float* outA = outX + (size_t)K_SEL * N_FEAT;       // 4096 x 4096

  xk_kernel  <<<N_NODES, 256, 0, stream>>>(X, kern, t);
  av_kernel  <<<N_NODES, 256, 0, stream>>>(A, t, y);
  rank_kernel<<<N_NODES / 256, 256, 0, stream>>>(y, sel);
  scan_kernel<<<1, 1024, 0, stream>>>(sel, idxb);
  xpool_kernel<<<K_SEL, 256, 0, stream>>>(X, y, idxb, outX);

  dim3 g(K_SEL / TN, K_SEL / TM);
  apool_gemm <<<g, 256, 0, stream>>>(A, idxb, outA);
}